// RBF_Layer_11321533792915
// MI455X (gfx1250) — compile-verified
//
#include <hip/hip_runtime.h>
#include <hip/hip_bf16.h>

typedef __attribute__((ext_vector_type(2))) float v2f;
typedef __attribute__((ext_vector_type(8))) float v8f;

#define IN_F 3
#define OUT_F 512
#define TILE 16
#define WAVES_PER_BLOCK 8

// D = A x B + 0 via V_WMMA_F32_16X16X4_F32, wave32.
// A (16x4 f32): lane m=l%16; VGPR0 = K=(l/16)*2, VGPR1 = K=(l/16)*2+1
// B (4x16 f32): lane n=l%16; VGPR0 = row K=(l/16)*2, VGPR1 = row K=(l/16)*2+1
// C/D (16x16 f32): lane n=l%16; VGPR v = row M = v + 8*(l/16)
//
// Homogeneous-coordinate fusion:
//   A row m = [-2x0, -2x1, -2x2, ||x_m||^2]
//   B col n = [ c0,   c1,   c2,  1        ]
// => D[m][n] = ||x_m||^2 - 2<x_m,c_n>;  epilogue adds ||c_n||^2 (pk_add),
//    clamps, and evaluates exp(-sq/sigma^2) as native v_exp_f32 with the
//    (-log2e/sigma^2) factor folded into one per-tile scalar.
__global__ __launch_bounds__(WAVES_PER_BLOCK * 32)
void rbf_wmma_kernel(const float* __restrict__ x,
                     const float* __restrict__ centres,
                     const float* __restrict__ log_sigmas,
                     float* __restrict__ out)
{
    const int lane = threadIdx.x & 31;
    const int wave = threadIdx.x >> 5;
    const int half = lane >> 4;   // 0: lanes 0-15, 1: lanes 16-31
    const int n16  = lane & 15;

    const int row_base = (blockIdx.x * WAVES_PER_BLOCK + wave) * TILE;

    // ---- A fragment ----
    const int m = row_base + n16;
    const float x0 = x[m * IN_F + 0];
    const float x1 = x[m * IN_F + 1];
    const float x2 = x[m * IN_F + 2];
    const float xsq = fmaf(x0, x0, fmaf(x1, x1, x2 * x2));
    v2f a;
    a.x = (half == 0) ? (-2.0f * x0) : (-2.0f * x2);  // K = 0 or 2
    a.y = (half == 0) ? (-2.0f * x1) : xsq;           // K = 1 or 3 (homog)

    // Lane's output column within each tile is n16; rows are half*8 + v.
    float* const orow = out + (size_t)(row_base + half * 8) * OUT_F + n16;

    for (int t = 0; t < OUT_F / TILE; ++t) {
        const int col = t * TILE + n16;

        // ---- B fragment ----
        const float c0 = centres[col * IN_F + 0];
        const float c1 = centres[col * IN_F + 1];
        const float c2 = centres[col * IN_F + 2];
        const float csq = fmaf(c0, c0, fmaf(c1, c1, c2 * c2));
        v2f b;
        b.x = (half == 0) ? c0 : c2;    // K = 0 or 2
        b.y = (half == 0) ? c1 : 1.0f;  // K = 1 or 3 (homog)

        v8f acc = {};
        // acc = ||x||^2 - 2<x,c>
        acc = __builtin_amdgcn_wmma_f32_16x16x4_f32(
            /*neg_a=*/false, a, /*neg_b=*/false, b,
            /*c_mod=*/(short)0, acc, /*reuse_a=*/false, /*reuse_b=*/false);

        // s = -log2(e)/sigma^2 : exp(-sq/sigma^2) == exp2(sq * s)
        const float s = -1.44269504088896340736f * __expf(-2.0f * log_sigmas[col]);
        float* op = orow + (size_t)t * TILE;
#pragma unroll
        for (int v = 0; v < 8; ++v) {
            float sq = acc[v] + csq;          // full squared distance
            sq = fmaxf(sq, 0.0f);             // reference clamp
            op[(size_t)v * OUT_F] = __builtin_amdgcn_exp2f(sq * s);  // v_exp_f32
        }
    }
}

extern "C" void kernel_launch(void* const* d_in, const int* in_sizes, int n_in,
                              void* d_out, int out_size, void* d_ws, size_t ws_size,
                              hipStream_t stream) {
    const float* x       = (const float*)d_in[0];
    const float* centres = (const float*)d_in[1];
    const float* lsig    = (const float*)d_in[2];
    float* out           = (float*)d_out;

    const int n_rows = in_sizes[0] / IN_F;                  // 524288
    const int blocks = n_rows / (TILE * WAVES_PER_BLOCK);   // 4096
    rbf_wmma_kernel<<<blocks, WAVES_PER_BLOCK * 32, 0, stream>>>(x, centres, lsig, out);
}